// AdaptiveGraphPooling_33861522162215
// MI455X (gfx1250) — compile-verified
//
#include <hip/hip_runtime.h>
#include <hip/hip_bf16.h>
#include <math.h>

#define DD 128            // feature dim
#define AA 64             // attention dim
#define TILE 16           // node tile (WMMA M)
#define XSTR 132          // LDS row stride for x tile (bank-conflict-free WMMA A reads)
#define NEG_HUGE (-3.402823466e38f)

typedef __attribute__((ext_vector_type(2))) float v2f;
typedef __attribute__((ext_vector_type(8))) float v8f;

// ---------------------------------------------------------------------------
// Build segment starts[0..G] from the *sorted* batch array. Each boundary node
// fills the (small) gap of empty graphs behind it. No atomics, no fixup pass.
// ---------------------------------------------------------------------------
__global__ __launch_bounds__(256) void k_bounds(const long long* __restrict__ batch,
                                                int* __restrict__ starts, int N, int G) {
  int i = blockIdx.x * blockDim.x + threadIdx.x;
  if (i >= N) return;
  int b = (int)batch[i];
  if (i == 0) {
    for (int g = 0; g <= b; ++g) starts[g] = 0;
  } else {
    int pb = (int)batch[i - 1];
    for (int g = pb + 1; g <= b; ++g) starts[g] = i;
  }
  if (i == N - 1) {
    for (int g = b + 1; g <= G; ++g) starts[g] = N;
  }
}

// ---------------------------------------------------------------------------
// Main fused kernel: one workgroup (128 threads = 4 waves) per graph.
// Single pass over x with online softmax. WMMA B fragments (att_W) live in
// registers for the whole kernel; x tile goes registers -> LDS (for WMMA A)
// while mean/max/weighted-sum accumulate straight from the registers.
// ---------------------------------------------------------------------------
__global__ __launch_bounds__(128) void k_main(const float* __restrict__ x,
                                              const int* __restrict__ starts,
                                              const float* __restrict__ att_W,
                                              const float* __restrict__ att_b,
                                              const float* __restrict__ att_ctx,
                                              float* __restrict__ mean_ws,
                                              float* __restrict__ max_ws,
                                              float* __restrict__ attn_ws) {
  __shared__ float xt[TILE * XSTR];    // x tile, padded rows (8.25 KB); reused as 3x[4][128] at end
  __shared__ float sc[TILE];           // per-node scores
  __shared__ float ev[TILE];           // per-node exp(s - m)
  __shared__ float bc[4];              // [0]=m  [1]=z  [2]=scale

  const int g = blockIdx.x;
  const int t = threadIdx.x;           // 0..127
  const int s = starts[g];
  const int e = starts[g + 1];
  const int cnt = e - s;

  const int lane = t & 31;
  const int half = lane >> 4;          // WMMA half: K pair select / row M vs M+8
  const int ln   = lane & 15;
  const int wv   = t >> 5;             // wave id 0..3
  const int nb   = wv * 16;            // this wave's 16-column chunk of the 64 attn dims
  const float myb = att_b[nb + ln];
  const float myc = att_ctx[nb + ln];

  // ---- hoist B fragments (att_W) into registers: 32 x v2f per lane ----
  v2f bf[32];
  #pragma unroll
  for (int kk = 0; kk < 32; ++kk) {
    const int ka = 4 * kk + 2 * half;
    bf[kk].x = att_W[ka * AA + nb + ln];
    bf[kk].y = att_W[(ka + 1) * AA + nb + ln];
  }

  // transposed pooling ownership: thread (wv, c) owns cols 4c..4c+3,
  // rows {wv, wv+4, wv+8, wv+12} of each tile
  const int c4 = 4 * (t & 31);
  float fsum4[4] = {0.f, 0.f, 0.f, 0.f};
  float fmax4[4] = {NEG_HUGE, NEG_HUGE, NEG_HUGE, NEG_HUGE};
  float S4[4]   = {0.f, 0.f, 0.f, 0.f};

  if (t == 0) { bc[0] = NEG_HUGE; bc[1] = 0.f; }
  __syncthreads();

  for (int base = s; base < e; base += TILE) {
    const int nv = min(TILE, e - base);

    // ---- load x tile into registers (coalesced b128) and stage to LDS ----
    float4 stash[4];
    #pragma unroll
    for (int i = 0; i < 4; ++i) {
      const int n = wv + 4 * i;
      float4 v = make_float4(0.f, 0.f, 0.f, 0.f);
      if (n < nv) v = ((const float4*)x)[(size_t)(base + n) * (DD / 4) + (t & 31)];
      stash[i] = v;
      float* dst = &xt[n * XSTR + c4];
      dst[0] = v.x; dst[1] = v.y; dst[2] = v.z; dst[3] = v.w;
    }
    if (t < TILE) sc[t] = 0.f;
    __syncthreads();

    // ---- WMMA: y[16 x 64] = xt @ att_W ; this wave does cols [nb, nb+16) ----
    v8f cacc = {0.f, 0.f, 0.f, 0.f, 0.f, 0.f, 0.f, 0.f};
    #pragma unroll
    for (int kk = 0; kk < 32; ++kk) {
      const int ka = 4 * kk + 2 * half;
      v2f a;
      a.x = xt[ln * XSTR + ka];
      a.y = xt[ln * XSTR + ka + 1];
      cacc = __builtin_amdgcn_wmma_f32_16x16x4_f32(false, a, false, bf[kk], (short)0,
                                                   cacc, false, false);
    }

    // ---- scores: p = tanh(y + b[a]) * ctx[a]; reduce over the 16 a-lanes ----
    #pragma unroll
    for (int j = 0; j < 8; ++j) {
      float y = tanhf(cacc[j] + myb) * myc;   // (node = j + 8*half, a = nb + ln)
      y += __shfl_xor(y, 1);
      y += __shfl_xor(y, 2);
      y += __shfl_xor(y, 4);
      y += __shfl_xor(y, 8);
      if (ln == 0) atomicAdd(&sc[j + 8 * half], y);  // 4 waves combine their a-chunks
    }
    __syncthreads();

    // ---- online softmax bookkeeping (thread 0, 16 elements) ----
    if (t == 0) {
      float m = bc[0], z = bc[1];
      float tmax = NEG_HUGE;
      for (int n = 0; n < nv; ++n) tmax = fmaxf(tmax, sc[n]);
      float nm = fmaxf(m, tmax);
      float scale = expf(m - nm);             // first tile: exp(-huge) -> 0
      float zs = z * scale;
      for (int n = 0; n < nv; ++n) { float ee = expf(sc[n] - nm); ev[n] = ee; zs += ee; }
      for (int n = nv; n < TILE; ++n) ev[n] = 0.f;
      bc[0] = nm; bc[1] = zs; bc[2] = scale;
    }
    __syncthreads();

    // ---- pools + rescaled weighted accumulation from registers ----
    const float scl = bc[2];
    #pragma unroll
    for (int j = 0; j < 4; ++j) S4[j] *= scl;
    #pragma unroll
    for (int i = 0; i < 4; ++i) {
      const int n = wv + 4 * i;
      if (n < nv) {
        const float ee = ev[n];
        const float4 v = stash[i];
        fsum4[0] += v.x; fmax4[0] = fmaxf(fmax4[0], v.x); S4[0] += ee * v.x;
        fsum4[1] += v.y; fmax4[1] = fmaxf(fmax4[1], v.y); S4[1] += ee * v.y;
        fsum4[2] += v.z; fmax4[2] = fmaxf(fmax4[2], v.z); S4[2] += ee * v.z;
        fsum4[3] += v.w; fmax4[3] = fmaxf(fmax4[3], v.w); S4[3] += ee * v.w;
      }
    }
    // next-iteration barrier (after xt/sc stores) orders everything
  }

  // ---- cross-wave reduction of the 4 row-group partials (reuse xt as scratch)
  __syncthreads();
  float* sumb = xt;              // [4][128]
  float* maxb = xt + 512;        // [4][128]
  float* Sb   = xt + 1024;       // [4][128]
  #pragma unroll
  for (int j = 0; j < 4; ++j) {
    sumb[wv * DD + c4 + j] = fsum4[j];
    maxb[wv * DD + c4 + j] = fmax4[j];
    Sb[wv * DD + c4 + j]   = S4[j];
  }
  __syncthreads();

  float fsum = 0.f, fmax = NEG_HUGE, S = 0.f;
  #pragma unroll
  for (int h = 0; h < 4; ++h) {
    fsum += sumb[h * DD + t];
    fmax = fmaxf(fmax, maxb[h * DD + t]);
    S += Sb[h * DD + t];
  }

  const float z = bc[1];
  const size_t o = (size_t)g * DD + t;
  mean_ws[o] = fsum / fmaxf((float)cnt, 1.0f);
  max_ws[o]  = (cnt > 0) ? fmax : 0.0f;
  attn_ws[o] = S / (z + 1e-8f);
}

// ---------------------------------------------------------------------------
// Per-graph finish: attn_pool = attn_raw @ out_W + out_b; selector MLP;
// softmax over 3 strategies; blended output. One 128-thread block per graph.
// ---------------------------------------------------------------------------
__global__ __launch_bounds__(128) void k_final(const float* __restrict__ mean_ws,
                                               const float* __restrict__ max_ws,
                                               const float* __restrict__ attn_ws,
                                               const float* __restrict__ out_W,
                                               const float* __restrict__ out_b,
                                               const float* __restrict__ sel_W1,
                                               const float* __restrict__ sel_b1,
                                               const float* __restrict__ sel_W2,
                                               const float* __restrict__ sel_b2,
                                               float* __restrict__ out) {
  __shared__ float fl[3 * DD];   // [mean | max | attn_pool]
  __shared__ float araw[DD];
  __shared__ float h[DD];
  __shared__ float sw[3];

  const int g = blockIdx.x, t = threadIdx.x;
  const size_t o = (size_t)g * DD + t;

  const float mv = mean_ws[o];
  const float xv = max_ws[o];
  araw[t] = attn_ws[o];
  fl[t] = mv;
  fl[DD + t] = xv;
  __syncthreads();

  float acc = out_b[t];
  #pragma unroll 4
  for (int k = 0; k < DD; ++k) acc += araw[k] * out_W[k * DD + t];
  const float av = acc;
  fl[2 * DD + t] = av;
  __syncthreads();

  float hh = sel_b1[t];
  #pragma unroll 4
  for (int k = 0; k < 3 * DD; ++k) hh += fl[k] * sel_W1[k * DD + t];
  h[t] = fmaxf(hh, 0.f);
  __syncthreads();

  if (t < 3) {
    float l = sel_b2[t];
    for (int k = 0; k < DD; ++k) l += h[k] * sel_W2[k * 3 + t];
    sw[t] = l;
  }
  __syncthreads();
  if (t == 0) {
    float m = fmaxf(sw[0], fmaxf(sw[1], sw[2]));
    float e0 = expf(sw[0] - m), e1 = expf(sw[1] - m), e2 = expf(sw[2] - m);
    float iz = 1.f / (e0 + e1 + e2);
    sw[0] = e0 * iz; sw[1] = e1 * iz; sw[2] = e2 * iz;
  }
  __syncthreads();

  out[o] = mv * sw[0] + xv * sw[1] + av * sw[2];
}

// ---------------------------------------------------------------------------
extern "C" void kernel_launch(void* const* d_in, const int* in_sizes, int n_in,
                              void* d_out, int out_size, void* d_ws, size_t ws_size,
                              hipStream_t stream) {
  const float*     x       = (const float*)d_in[0];
  const long long* batch   = (const long long*)d_in[1];   // int64 in reference
  const float*     att_W   = (const float*)d_in[2];
  const float*     att_b   = (const float*)d_in[3];
  const float*     att_ctx = (const float*)d_in[4];
  const float*     out_W   = (const float*)d_in[5];
  const float*     out_b   = (const float*)d_in[6];
  const float*     sel_W1  = (const float*)d_in[7];
  const float*     sel_b1  = (const float*)d_in[8];
  const float*     sel_W2  = (const float*)d_in[9];
  const float*     sel_b2  = (const float*)d_in[10];
  float*           out     = (float*)d_out;

  const int N = in_sizes[0] / DD;
  const int G = out_size / DD;

  // workspace layout
  size_t off = 0;
  int* starts = (int*)((char*)d_ws + off);
  off += ((size_t)(G + 1) * sizeof(int) + 255) & ~(size_t)255;
  float* mean_ws = (float*)((char*)d_ws + off); off += (size_t)G * DD * sizeof(float);
  float* max_ws  = (float*)((char*)d_ws + off); off += (size_t)G * DD * sizeof(float);
  float* attn_ws = (float*)((char*)d_ws + off); off += (size_t)G * DD * sizeof(float);
  (void)ws_size; (void)n_in;

  k_bounds<<<(N + 255) / 256, 256, 0, stream>>>(batch, starts, N, G);
  k_main<<<G, 128, 0, stream>>>(x, starts, att_W, att_b, att_ctx,
                                mean_ws, max_ws, attn_ws);
  k_final<<<G, 128, 0, stream>>>(mean_ws, max_ws, attn_ws, out_W, out_b,
                                 sel_W1, sel_b1, sel_W2, sel_b2, out);
}